// AngularDescriptor_61847529062469
// MI455X (gfx1250) — compile-verified
//
#include <hip/hip_runtime.h>
#include <hip/hip_bf16.h>

// ---------------------------------------------------------------------------
// Angular descriptor (Chebyshev x Legendre) for MI455X / gfx1250.
// Phase 1: per-edge radial embedding + 8x8 matvec  -> workspace (SoA).
// Phase 2: TWO atoms per wave packed into one WMMA tile:
//            A rows 0..7  = U_atomA,  rows 8..15 = U_atomB
//            B cols 0..3  = P_atomA,  cols 4..7  = P_atomB
//            D(0..7,0..3) = Q_A,      D(8..15,4..7) = Q_B  (diag blocks)
//          K = pair index (shared by all atoms), 4 pairs per
//          V_WMMA_F32_16X16X4_F32. Branch-free operand fetch from LDS
//          (lanes m>=8 read a zero slot for B with stride 0) -> full EXEC,
//          no saveexec in the hot loop, loads merge to ds_load_2addr_b32.
// ---------------------------------------------------------------------------

typedef float v2f __attribute__((ext_vector_type(2)));
typedef float v8f __attribute__((ext_vector_type(8)));

#define KMAX  8
#define NDESC 8
#define LMAX  4
#define RC    5.0f
#define MAXM  20          // neighbor capacity (problem: M_NBR = 20)
#define MAXPP 192         // padded pair count: 4*ceil(190/4)
#define WPB   2           // waves per 64-thread block (each wave: 2 atoms)

// ------------------------- Phase 1: edge kernel ----------------------------
__global__ void AngDesc_edge_kernel(const int*   __restrict__ types,
                                    const float* __restrict__ pos,
                                    const int*   __restrict__ nbr,
                                    const float* __restrict__ ctab,
                                    float*       __restrict__ ws,
                                    int N, int M, int nTypes)
{
    int e  = blockIdx.x * blockDim.x + threadIdx.x;
    int NM = N * M;
    if (e >= NM) return;

    int i = e / M;
    int j = nbr[e];

    float xi = pos[3*i+0], yi = pos[3*i+1], zi = pos[3*i+2];
    float rx = pos[3*j+0] - xi;
    float ry = pos[3*j+1] - yi;
    float rz = pos[3*j+2] - zi;
    float r  = sqrtf(rx*rx + ry*ry + rz*rz);

    // cutoff + Chebyshev basis, matching reference exactly in fp32
    float fc = 0.0f;
    if (r < RC) fc = 0.5f * cosf(3.14159265358979323846f * r * (1.0f/RC)) + 0.5f;
    float hfc = 0.5f * fc;
    float t = r * (1.0f/RC) - 1.0f;
    float x = 2.0f * t * t - 1.0f;

    float f[KMAX];
    float Tm2 = 1.0f, Tm1 = x;
    f[0] = (Tm2 + 1.0f) * hfc;
    f[1] = (Tm1 + 1.0f) * hfc;
    #pragma unroll
    for (int k = 2; k < KMAX; ++k) {
        float Tk = 2.0f * x * Tm1 - Tm2;
        f[k] = (Tk + 1.0f) * hfc;
        Tm2 = Tm1; Tm1 = Tk;
    }

    int ti = types[i], tj = types[j];
    const float* c = ctab + (size_t)((ti * nTypes + tj) * NDESC) * KMAX;

    // SoA workspace: comp-major for coalesced stores; comps 0..3 = rx,ry,rz,r
    ws[0*NM + e] = rx;
    ws[1*NM + e] = ry;
    ws[2*NM + e] = rz;
    ws[3*NM + e] = r;
    #pragma unroll
    for (int d = 0; d < NDESC; ++d) {
        float g = 0.0f;
        #pragma unroll
        for (int k = 0; k < KMAX; ++k) g = fmaf(c[d*KMAX + k], f[k], g);
        ws[(4 + d)*NM + e] = g;
    }
}

// ------------------- Phase 2: per-wave 2-atom WMMA -------------------------
__global__ void __launch_bounds__(32 * WPB)
AngDesc_pair_kernel(float*       __restrict__ out,
                    const float* __restrict__ ws,
                    int N, int M)
{
    __shared__ int   pairTab[MAXPP];                   // (j | k<<8)
    __shared__ float E [WPB][2][12 * MAXM];            // edge stage, 2 atoms
    __shared__ float PT[WPB][MAXPP * 8 + 16];          // B: [p][2*4] + zeros
    __shared__ float UT[WPB][MAXPP * 16];              // A: [p][2*8] interleaved

    const int lane = threadIdx.x & 31;
    const int w    = threadIdx.x >> 5;
    const int m    = lane & 15;        // A: M-row / B,D: N-col owned by lane
    const int half = lane >> 4;        // selects K pair {0,1} vs {2,3}
    const bool m8  = (m < 8);

    const int P    = M * (M - 1) / 2;          // 190
    const int PPAD = ((P + 3) >> 2) << 2;      // 192
    const int NM   = N * M;

    // Triangular pair-index table, once per block (0 for padding pairs).
    for (int p = threadIdx.x; p < MAXPP; p += blockDim.x) {
        int val = 0;
        if (p < P) {
            int a = 0, rem = p;
            while (rem >= M - 1 - a) { rem -= (M - 1 - a); ++a; }
            val = a | ((a + 1 + rem) << 8);
        }
        pairTab[p] = val;
    }
    __syncthreads();

    const int stepAtoms = gridDim.x * WPB * 2;
    const int iters = (N + stepAtoms - 1) / stepAtoms;   // uniform trip count
    int atomA = (blockIdx.x * WPB + w) * 2;

    for (int it = 0; it < iters; ++it, atomA += stepAtoms) {
        const bool activeA = (atomA < N);
        const int  atomB   = atomA + 1;
        const bool activeB = (atomB < N);
        const int  atomBc  = activeB ? atomB : atomA;   // clamp (never stored)

        __syncthreads();                       // guard LDS reuse across iters
        if (activeA) {
            // Stage both atoms' edges: comps 0..3 = rvec,r ; 4..11 = g[0..7]
            int baseA = atomA * M, baseB = atomBc * M;
            #pragma unroll
            for (int comp = 0; comp < 12; ++comp) {
                if (lane < M) {
                    E[w][0][comp*M + lane] = ws[comp*NM + baseA + lane];
                    E[w][1][comp*M + lane] = ws[comp*NM + baseB + lane];
                }
            }
        }
        __syncthreads();

        if (activeA) {
            // Legendre table: PT[p*8 + s*4 + l], s = atom slot. Padded pairs
            // get all-zero B rows, killing their contribution.
            for (int p = lane; p < PPAD; p += 32) {
                float pl[8] = {0,0,0,0,0,0,0,0};
                if (p < P) {
                    int pk = pairTab[p];
                    int a = pk & 255, b = pk >> 8;
                    #pragma unroll
                    for (int s = 0; s < 2; ++s) {
                        const float* Ew = E[w][s];
                        float ca = (Ew[0*M+a]*Ew[0*M+b] +
                                    Ew[1*M+a]*Ew[1*M+b] +
                                    Ew[2*M+a]*Ew[2*M+b])
                                   / (Ew[3*M+a]*Ew[3*M+b] + 1e-8f);
                        pl[s*4+0] = 1.0f;
                        pl[s*4+1] = ca;
                        pl[s*4+2] = 0.5f * (3.0f*ca*ca - 1.0f);
                        pl[s*4+3] = (5.0f*ca*pl[s*4+2] - 2.0f*ca) * (1.0f/3.0f);
                    }
                }
                float* q = &PT[w][p * 8];
                #pragma unroll
                for (int n = 0; n < 8; ++n) q[n] = pl[n];
            }

            // U table: UT[p*16 + s*8 + d] = g_j[d]*g_k[d] for atom slot s.
            // Lane owns fixed (s,d); two lane-halves cover even/odd p.
            {
                int s = m >> 3, d = m & 7;
                const float* Gs = &E[w][s][4 * M];
                int off = d * M;
                for (int p = half; p < PPAD; p += 2) {
                    int pk = pairTab[p];
                    UT[w][p*16 + m] = Gs[off + (pk & 255)] * Gs[off + (pk >> 8)];
                }
            }

            // Zero slot for B lanes n>=8 (cols 8..15 unused).
            if (lane < 16) PT[w][PPAD * 8 + lane] = 0.0f;
        }
        __syncthreads();

        if (activeA) {
            // D[16x16] += A[16x4] x B[4x16]; K advances 4 pairs per WMMA.
            // All A lanes useful; B lanes m>=8 walk the zero slot, stride 0.
            const float* Ub = UT[w];
            const float* Pb = PT[w];
            int aIdx  = (half << 1) * 16 + m;                 // A(m, 2h), +16 -> 2h+1
            int bIdx  = m8 ? ((half << 1) * 8 + m) : (PPAD * 8);
            int bStep = m8 ? 32 : 0;                          // 4 pairs * 8 floats

            v8f acc = {};
            const int nwm = PPAD >> 2;          // 48
            #pragma unroll 4
            for (int s = 0; s < nwm; ++s) {
                v2f Av = { Ub[aIdx], Ub[aIdx + 16] };   // ds_load_2addr off1:16
                v2f Bv = { Pb[bIdx], Pb[bIdx + 8]  };   // ds_load_2addr off1:8
                acc = __builtin_amdgcn_wmma_f32_16x16x4_f32(
                        false, Av, false, Bv, (short)0, acc, false, false);
                aIdx += 64;
                bIdx += bStep;
            }

            // D layout: VGPR v, lanes 0..15 -> (M=v,   N=lane),
            //                    lanes 16..31 -> (M=v+8, N=lane-16).
            // Q_A = D(0..7,0..3): lanes 0..3.  Q_B = D(8..15,4..7): lanes 20..23.
            if (lane < LMAX) {
                float* o = out + (size_t)atomA * (NDESC * LMAX);
                #pragma unroll
                for (int v = 0; v < NDESC; ++v) o[v * LMAX + lane] = acc[v];
            }
            if (activeB && lane >= 20 && lane < 20 + LMAX) {
                float* o = out + (size_t)atomB * (NDESC * LMAX) + (lane - 20);
                #pragma unroll
                for (int v = 0; v < NDESC; ++v) o[v * LMAX] = acc[v];
            }
        }
    }
}

// ------------------------------- launcher ----------------------------------
extern "C" void kernel_launch(void* const* d_in, const int* in_sizes, int n_in,
                              void* d_out, int out_size, void* d_ws, size_t ws_size,
                              hipStream_t stream) {
    const int*   types = (const int*)  d_in[0];
    const float* pos   = (const float*)d_in[1];
    const int*   nbr   = (const int*)  d_in[2];
    const float* ctab  = (const float*)d_in[3];
    float* ws  = (float*)d_ws;
    float* out = (float*)d_out;

    int N = in_sizes[0];
    int M = in_sizes[2] / N;                       // 20
    int nTypes = 1;
    while (nTypes * nTypes * NDESC * KMAX < in_sizes[3]) ++nTypes;   // 4

    int NM = N * M;
    int eb = (NM + 255) / 256;
    AngDesc_edge_kernel<<<eb, 256, 0, stream>>>(types, pos, nbr, ctab, ws,
                                                N, M, nTypes);

    int atomsPerBlock = WPB * 2;                   // 2 waves x 2 atoms
    int pb = (N + atomsPerBlock - 1) / atomsPerBlock;
    AngDesc_pair_kernel<<<pb, 32 * WPB, 0, stream>>>(out, ws, N, M);
}